// NTXentLoss_poly_43568148251256
// MI455X (gfx1250) — compile-verified
//
#include <hip/hip_runtime.h>
#include <hip/hip_bf16.h>
#include <math.h>

// NT-Xent (SimCLR) loss, fused flash-style on gfx1250.
// reps = concat(zjs, zis) [4096,256] f32; sim = rn @ rn^T computed tile-wise
// with V_WMMA_F32_16X16X4_F32 (native f32 WMMA -> matches f32 reference
// precision); per-row softmax stats accumulated on the fly (no 64MB sim
// matrix). All reductions are fixed-order (shuffles + serial sums) so the
// result is bitwise deterministic across replays.
//
// Round-3 change: fully branch-free tile epilogue. The positive similarity
// is now accumulated as a masked register sum (it occurs exactly once per
// row, so sum == value) instead of a divergent ds_store — removes the 8x
// s_and_saveexec/ds_store/exec-restore blocks per tile seen in round-2 asm.

typedef __attribute__((ext_vector_type(2))) float v2f;
typedef __attribute__((ext_vector_type(8))) float v8f;

#define TWO_N  4096
#define HALF_N 2048
#define DIM    256
#define INV_T  5.0f        // 1 / 0.2
#define EPS_N  1e-8f
#define A_STRIDE 258       // 16x256 tile padded to kill LDS bank conflicts

// ---------------------------------------------------------------------------
// Kernel 1: rn[row] = reps[row] / max(||reps[row]||, eps)
// rows 0..2047 <- zjs, rows 2048..4095 <- zis (reference concat order!)
// ---------------------------------------------------------------------------
__global__ __launch_bounds__(256)
void ntxent_normalize(const float* __restrict__ zis,
                      const float* __restrict__ zjs,
                      float* __restrict__ rn) {
  __shared__ float red[256];
  __shared__ float inv;
  const int row = blockIdx.x;
  const int t = threadIdx.x;
  const float* src = (row < HALF_N) ? (zjs + (size_t)row * DIM)
                                    : (zis + (size_t)(row - HALF_N) * DIM);
  const float x = src[t];
  red[t] = x * x;
  __syncthreads();
  for (int s = 128; s > 0; s >>= 1) {
    if (t < s) red[t] += red[t + s];
    __syncthreads();
  }
  if (t == 0) {
    float nrm = fmaxf(sqrtf(red[0]), EPS_N);
    inv = 1.0f / nrm;
  }
  __syncthreads();
  rn[(size_t)row * DIM + t] = x * inv;
}

// ---------------------------------------------------------------------------
// Kernel 2: fused sim-row pass. Block b owns rows i0 = 16*b .. i0+15.
// 8 waves per block; wave w streams column tiles jt = w, w+8, ... (32 tiles).
// For each 16x16 tile: 64x V_WMMA_F32_16X16X4_F32 over K=256.
//
// f32 WMMA fragment layout (ISA 7.12.2):
//   A (16x4): lane L -> row (L&15), K = k0 + 2*(L>>4) + {v0,v1}  -> float2
//   B (4x16): lane L -> col (L&15), K = k0 + 2*(L>>4) + {v0,v1}  -> float2
//   (B tile is R^T, so lane L reads row j0+(L&15) of rn -> same load shape)
//   D (16x16): vgpr r, lane L -> M = r + 8*(L>>4), N = (L&15)
// ---------------------------------------------------------------------------
__global__ __launch_bounds__(256)
void ntxent_fused(const float* __restrict__ rn,
                  float* __restrict__ partials) {
  __shared__ alignas(16) float Alds[16 * A_STRIDE];
  __shared__ float wsum[8][16];   // per-wave exp-sums per row
  __shared__ float wpos[8][16];   // per-wave positive partials per row
  __shared__ float cearr[16];
  __shared__ float prarr[16];

  const int i0    = blockIdx.x * 16;
  const int tid   = threadIdx.x;
  const int wave  = tid >> 5;
  const int lane  = tid & 31;
  const int lrow  = lane & 15;          // A row-in-tile / B col-in-tile
  const int khalf = (lane >> 4) * 2;    // K sub-offset for upper half-wave
  const int mbase = (lane >> 4) * 8;    // D rows this lane holds

  // Stage A tile (16 rows x 256 f32) into LDS, coalesced.
  for (int idx = tid; idx < 16 * DIM; idx += 256) {
    const int r = idx >> 8;
    const int k = idx & (DIM - 1);
    Alds[r * A_STRIDE + k] = rn[(size_t)(i0 + r) * DIM + k];
  }
  __syncthreads();

  float eacc[8];   // sum of exp(s/T) over negatives
  float pacc[8];   // positive value (occurs exactly once -> masked sum)
#pragma unroll
  for (int r = 0; r < 8; ++r) { eacc[r] = 0.0f; pacc[r] = 0.0f; }

  const float* Aptr = &Alds[lrow * A_STRIDE + khalf];

  for (int jt = wave; jt < TWO_N / 16; jt += 8) {
    const int j0 = jt * 16;
    const float* Brow = rn + (size_t)(j0 + lrow) * DIM + khalf;

    v8f acc = {0.f, 0.f, 0.f, 0.f, 0.f, 0.f, 0.f, 0.f};
#pragma unroll 8
    for (int k0 = 0; k0 < DIM; k0 += 4) {
      v2f a = *(const v2f*)(Aptr + k0);
      v2f b = *(const v2f*)(Brow + k0);
      // 8 args: (neg_a, A, neg_b, B, c_mod, C, reuse_a, reuse_b)
      acc = __builtin_amdgcn_wmma_f32_16x16x4_f32(
          false, a, false, b, (short)0, acc, false, false);
    }

    const int j = j0 + lrow;
#pragma unroll
    for (int r = 0; r < 8; ++r) {
      const int i = i0 + mbase + r;
      const float s = acc[r];
      const bool isDiag = (j == i);
      const bool isPos  = (j == (i ^ HALF_N));
      // Fully branch-free: v_exp_f32 always; v_cndmask selects contribution.
      const float e = __expf(s * INV_T);
      eacc[r] += (isDiag || isPos) ? 0.0f : e;
      pacc[r] += isPos ? s : 0.0f;
    }
  }

  // Deterministic wave-level reduction across the 16 lanes sharing each row
  // set (xor offsets 1,2,4,8 stay within each 16-lane half).
#pragma unroll
  for (int r = 0; r < 8; ++r) {
    float v = eacc[r];
    v += __shfl_xor(v, 1, 32);
    v += __shfl_xor(v, 2, 32);
    v += __shfl_xor(v, 4, 32);
    v += __shfl_xor(v, 8, 32);
    eacc[r] = v;
    float p = pacc[r];
    p += __shfl_xor(p, 1, 32);
    p += __shfl_xor(p, 2, 32);
    p += __shfl_xor(p, 4, 32);
    p += __shfl_xor(p, 8, 32);
    pacc[r] = p;
  }
  if (lrow == 0) {                  // lanes 0 (rows 0-7) and 16 (rows 8-15)
#pragma unroll
    for (int r = 0; r < 8; ++r) {
      wsum[wave][mbase + r] = eacc[r];
      wpos[wave][mbase + r] = pacc[r];
    }
  }
  __syncthreads();

  if (tid < 16) {
    float se = 0.f, p = 0.f;
    for (int w = 0; w < 8; ++w) {     // fixed order
      se += wsum[w][tid];
      p  += wpos[w][tid];             // all-but-one terms are exact zeros
    }
    const float pe = expf(p * INV_T);
    const float denom = pe + se;
    cearr[tid] = logf(denom) - p * INV_T;   // -log_softmax[0]
    prarr[tid] = pe / denom;                // softmax prob of the positive
  }
  __syncthreads();

  if (tid == 0) {
    float ce = 0.f, pr = 0.f;
    for (int t = 0; t < 16; ++t) { ce += cearr[t]; pr += prarr[t]; }
    partials[blockIdx.x * 2 + 0] = ce;
    partials[blockIdx.x * 2 + 1] = pr;
  }
}

// ---------------------------------------------------------------------------
// Kernel 3: fixed-order reduction of 256 block partials -> scalar loss.
// loss = CE/4096 + N*(1/N - pt)  with  pt = sum(probs0)/(4096*4095)
//      = CE/4096 + 1 - sum(probs0)/8190
// ---------------------------------------------------------------------------
__global__ void ntxent_finalize(const float* __restrict__ partials,
                                float* __restrict__ out) {
  if (threadIdx.x == 0 && blockIdx.x == 0) {
    float ce = 0.f, pr = 0.f;
    for (int b = 0; b < TWO_N / 16; ++b) {
      ce += partials[2 * b + 0];
      pr += partials[2 * b + 1];
    }
    out[0] = ce / (float)TWO_N + 1.0f - pr / (float)(2 * (TWO_N - 1));
  }
}

extern "C" void kernel_launch(void* const* d_in, const int* in_sizes, int n_in,
                              void* d_out, int out_size, void* d_ws, size_t ws_size,
                              hipStream_t stream) {
  const float* zis = (const float*)d_in[0];   // [2048, 256] f32
  const float* zjs = (const float*)d_in[1];   // [2048, 256] f32

  float* rn       = (float*)d_ws;                    // [4096*256] f32 = 4MB
  float* partials = rn + (size_t)TWO_N * DIM;        // [256*2] f32

  ntxent_normalize<<<TWO_N, 256, 0, stream>>>(zis, zjs, rn);
  ntxent_fused<<<TWO_N / 16, 256, 0, stream>>>(rn, partials);
  ntxent_finalize<<<1, 32, 0, stream>>>(partials, (float*)d_out);
}